// ExplicitComplexityFusion_18631568130348
// MI455X (gfx1250) — compile-verified
//
#include <hip/hip_runtime.h>
#include <hip/hip_bf16.h>
#include <stdint.h>

// Problem constants (match the reference).
#define N_NODES    200000
#define D          256
#define N_EDGES    6400000
#define BATCH_SIZE 512
#define TILE_ROWS  16
#define N_TILES    (N_NODES / TILE_ROWS)   // 12500, exact
#define FUSE_BLOCK 256                      // 8 wave32 waves

typedef float v4f __attribute__((ext_vector_type(4)));
typedef int   v4i __attribute__((ext_vector_type(4)));

// gfx1250 async global->LDS path, feature-detected so we compile cleanly on
// both ROCm 7.2 (clang-22) and amdgpu-toolchain (clang-23).
#if defined(__gfx1250__) &&                                              \
    __has_builtin(__builtin_amdgcn_global_load_async_to_lds_b128) &&     \
    __has_builtin(__builtin_amdgcn_s_wait_asynccnt)
#define USE_ASYNC_LDS 1
#else
#define USE_ASYNC_LDS 0
#endif

#if USE_ASYNC_LDS
typedef __attribute__((address_space(1))) v4i* gv4i_ptr;  // global int4*
typedef __attribute__((address_space(3))) v4i* lv4i_ptr;  // LDS int4*
#endif

__device__ __forceinline__ void async_copy_b128(const float* g, float* l) {
#if USE_ASYNC_LDS
  // Signature: (int4 addrspace(1)* gaddr, int4 addrspace(3)* ldsaddr,
  //             imm offset, imm cpol) -> global_load_async_to_lds_b128
  __builtin_amdgcn_global_load_async_to_lds_b128(
      (gv4i_ptr)(g), (lv4i_ptr)(l), /*offset=*/0, /*cpol=*/0);
#else
  (void)g; (void)l;
#endif
}

__device__ __forceinline__ void wait_async_zero() {
#if USE_ASYNC_LDS
  __builtin_amdgcn_s_wait_asynccnt(0);
#endif
}

// ---------------------------------------------------------------------------
// Kernel 1: per-graph node counts via binary search on the sorted batch array,
// and zero the edge-count scratch (d_ws is poisoned by the harness).
// One block of 512 threads; ~36 L2-resident loads per thread.
// ---------------------------------------------------------------------------
__global__ void counts_kernel(const int* __restrict__ batch,
                              int* __restrict__ counts,
                              int* __restrict__ n_edges) {
  const int b = threadIdx.x;  // 0..511
  int lo = 0, hi = N_NODES;
  while (lo < hi) {           // lower_bound(b)
    const int mid = (lo + hi) >> 1;
    if (batch[mid] < b) lo = mid + 1; else hi = mid;
  }
  const int start = lo;
  hi = N_NODES;
  while (lo < hi) {           // lower_bound(b+1)
    const int mid = (lo + hi) >> 1;
    if (batch[mid] < b + 1) lo = mid + 1; else hi = mid;
  }
  counts[b]  = lo - start;
  n_edges[b] = 0;
}

// ---------------------------------------------------------------------------
// Kernel 2: intra-graph edge counts. 6.4M edges, LDS histogram of 512 bins,
// flush with one global atomic per bin per block (512K atomics total instead
// of 6.4M). batch[] gathers hit the 192MB L2 (table is 800KB).
// ---------------------------------------------------------------------------
__global__ __launch_bounds__(256)
void edge_count_kernel(const int* __restrict__ ei,
                       const int* __restrict__ batch,
                       int* __restrict__ n_edges) {
  __shared__ int h[BATCH_SIZE];
  for (int i = threadIdx.x; i < BATCH_SIZE; i += blockDim.x) h[i] = 0;
  __syncthreads();

  const int stride = gridDim.x * blockDim.x;
  for (int e = blockIdx.x * blockDim.x + threadIdx.x; e < N_EDGES; e += stride) {
    const int s  = ei[e];              // row 0 of edge_index (coalesced)
    const int d  = ei[e + N_EDGES];    // row 1 of edge_index (coalesced)
    const int bs = batch[s];           // L2-resident gather
    const int bd = batch[d];
    if (bs == bd) atomicAdd(&h[bs], 1);  // ds_add_u32
  }
  __syncthreads();
  for (int i = threadIdx.x; i < BATCH_SIZE; i += blockDim.x) {
    const int c = h[i];
    if (c) atomicAdd(&n_edges[i], c);
  }
}

// ---------------------------------------------------------------------------
// Kernel 3: per-graph weights. Writes w_local / w_global straight into the
// output tail (tuple layout: x_fused | w_local | w_global).
// ---------------------------------------------------------------------------
__global__ void weights_kernel(const int* __restrict__ counts,
                               const int* __restrict__ n_edges,
                               const float* __restrict__ alpha,
                               const float* __restrict__ beta,
                               const float* __restrict__ gamma,
                               float* __restrict__ out) {
  const int b = threadIdx.x;  // 0..511
  const float nn = (float)counts[b];
  const float ne = (float)n_edges[b];
  const float scale   = logf(nn + 1.0f) / logf(501.0f);  // log(MAX_NODES+1)
  const float density = ne / (nn * (nn - 1.0f) + 1e-8f);
  const float z  = alpha[0] * scale + beta[0] * density + gamma[0];
  const float wg = 1.0f / (1.0f + expf(-z));
  float* wl_out = out + (size_t)N_NODES * D;
  wl_out[b]              = 1.0f - wg;  // w_local
  wl_out[BATCH_SIZE + b] = wg;         // w_global
}

// ---------------------------------------------------------------------------
// Kernel 4: fused blend. One block per 16-row tile (16x256 f32 = 16KB per
// input matrix). gfx1250 path: async global->LDS b128 copies (ASYNCcnt),
// s_wait_asynccnt 0 + barrier, compute from LDS, non-temporal b128 stores so
// the 204MB output stream doesn't rinse L2. Fallback: NT loads direct.
// ---------------------------------------------------------------------------
__global__ __launch_bounds__(FUSE_BLOCK)
void fuse_kernel(const float* __restrict__ xg,
                 const float* __restrict__ xa,
                 const int* __restrict__ batch,
                 float* __restrict__ out) {
  __shared__ float ldsG[TILE_ROWS * D];   // 16 KB
  __shared__ float ldsA[TILE_ROWS * D];   // 16 KB
  __shared__ float s_wl[TILE_ROWS];
  __shared__ float s_wg[TILE_ROWS];

  const int    tid = threadIdx.x;
  const size_t r0  = (size_t)blockIdx.x * TILE_ROWS;
  const float* gG  = xg + r0 * D;
  const float* gA  = xa + r0 * D;
  const float* wl_g = out + (size_t)N_NODES * D;   // written by weights_kernel
  const float* wg_g = wl_g + BATCH_SIZE;

  // 1024 float4 per matrix per tile; 4 per thread.
#if USE_ASYNC_LDS
  #pragma unroll
  for (int k = 0; k < (TILE_ROWS * D / 4) / FUSE_BLOCK; ++k) {
    const int i = tid + k * FUSE_BLOCK;         // float4 index within tile
    async_copy_b128(gG + i * 4, &ldsG[i * 4]);  // global_load_async_to_lds_b128
    async_copy_b128(gA + i * 4, &ldsA[i * 4]);
  }
#endif

  if (tid < TILE_ROWS) {
    const int b = batch[r0 + tid];
    s_wl[tid] = wl_g[b];
    s_wg[tid] = wg_g[b];
  }

  wait_async_zero();   // s_wait_asynccnt 0 (our own async writes to LDS)
  __syncthreads();     // make all waves' LDS writes visible

  #pragma unroll
  for (int k = 0; k < (TILE_ROWS * D / 4) / FUSE_BLOCK; ++k) {
    const int i   = tid + k * FUSE_BLOCK;
    const int row = i >> 6;                     // 64 float4 per 256-wide row
#if USE_ASYNC_LDS
    const v4f g = *(const v4f*)&ldsG[i * 4];    // ds_load_b128
    const v4f a = *(const v4f*)&ldsA[i * 4];
#else
    const v4f g = __builtin_nontemporal_load((const v4f*)(gG + i * 4));
    const v4f a = __builtin_nontemporal_load((const v4f*)(gA + i * 4));
#endif
    const v4f f = s_wl[row] * g + s_wg[row] * a;
    __builtin_nontemporal_store(f, (v4f*)(out + r0 * D + i * 4));  // TH=NT
  }
}

// ---------------------------------------------------------------------------
extern "C" void kernel_launch(void* const* d_in, const int* in_sizes, int n_in,
                              void* d_out, int out_size, void* d_ws, size_t ws_size,
                              hipStream_t stream) {
  (void)in_sizes; (void)n_in; (void)out_size; (void)ws_size;
  const float* xg    = (const float*)d_in[0];  // x_ggnn  [200000*256]
  const float* xa    = (const float*)d_in[1];  // x_appnp [200000*256]
  const int*   ei    = (const int*)d_in[2];    // edge_index [2*6400000]
  const int*   batch = (const int*)d_in[3];    // batch [200000], sorted
  const float* alpha = (const float*)d_in[4];
  const float* beta  = (const float*)d_in[5];
  const float* gamma = (const float*)d_in[6];
  float* out = (float*)d_out;

  int* counts  = (int*)d_ws;            // [512]
  int* n_edges = counts + BATCH_SIZE;   // [512]

  counts_kernel<<<1, BATCH_SIZE, 0, stream>>>(batch, counts, n_edges);
  edge_count_kernel<<<1024, 256, 0, stream>>>(ei, batch, n_edges);
  weights_kernel<<<1, BATCH_SIZE, 0, stream>>>(counts, n_edges, alpha, beta, gamma, out);
  fuse_kernel<<<N_TILES, FUSE_BLOCK, 0, stream>>>(xg, xa, batch, out);
}